// NFQuantizer_82798379532948
// MI455X (gfx1250) — compile-verified
//
#include <hip/hip_runtime.h>

// NF4 quantize + straight-through forward for [rows x 8192] fp32.
// Memory-bound: 512 MiB min traffic -> ~22us @ 23.3 TB/s on MI455X.
// CDNA5 path: row staged to LDS with global_load_async_to_lds_b128
// (ASYNCcnt), drained with s_wait_asynccnt, consumed via ds_load_b128.
// searchsorted = explicit 4-step binary search with LDS tables (avoids
// clang rewriting select trees into private scratch arrays).

#define COLS   8192
#define TPB    256
#define CHUNKS (COLS / (TPB * 4))   // 8 float4 chunks per thread

__global__ __launch_bounds__(TPB)
void nf4_quant_row_kernel(const float* __restrict__ x,
                          const float* __restrict__ bnd,
                          const float* __restrict__ dt,
                          float* __restrict__ out)
{
    __shared__ float tile[COLS];          // 32 KB row stage
    __shared__ float bnd_s[16];           // 15 boundaries (+1 pad)
    __shared__ float dt_s[16];            // 16 NF4 levels
    __shared__ float red[TPB / 32];       // per-wave maxima

    const int tid = threadIdx.x;
    const int row = blockIdx.x;

    // stage tiny tables into LDS (covered by the barrier below)
    if (tid < 16) {
        dt_s[tid]  = dt[tid];
        bnd_s[tid] = (tid < 15) ? bnd[tid] : 0.0f;
    }
    const float b7 = bnd[7];              // uniform -> SGPR, used in v_cmp

    const float* rowp = x + (size_t)row * COLS;
    const unsigned lds_base = (unsigned)(unsigned long long)(&tile[0]);

    // ---- async copy: HBM -> LDS, 16 B per lane per instruction ----
    #pragma unroll
    for (int k = 0; k < CHUNKS; ++k) {
        unsigned goff = (unsigned)((tid + k * TPB) * 16);   // byte offset in row
        unsigned loff = lds_base + goff;                    // LDS byte address
        asm volatile("global_load_async_to_lds_b128 %0, %1, %2"
                     :: "v"(loff), "v"(goff), "s"(rowp)
                     : "memory");
    }
    asm volatile("s_wait_asynccnt 0x0" ::: "memory");
    __syncthreads();   // all waves' async stores + table staging visible

    // ---- pass 1: per-thread absmax over its 8 float4 chunks ----
    const float4* t4 = reinterpret_cast<const float4*>(tile);
    float m = 0.0f;
    #pragma unroll
    for (int k = 0; k < CHUNKS; ++k) {
        float4 v = t4[tid + k * TPB];
        m = fmaxf(m, fmaxf(fmaxf(fabsf(v.x), fabsf(v.y)),
                           fmaxf(fabsf(v.z), fabsf(v.w))));
    }

    // wave32 butterfly reduction, then cross-wave via LDS
    #pragma unroll
    for (int off = 16; off > 0; off >>= 1)
        m = fmaxf(m, __shfl_xor(m, off, 32));
    if ((tid & 31) == 0) red[tid >> 5] = m;
    __syncthreads();
    float rm = red[0];
    #pragma unroll
    for (int w = 1; w < TPB / 32; ++w) rm = fmaxf(rm, red[w]);
    const float s = fmaxf(rm, 1e-6f);

    // ---- pass 2: quantize + coalesced b128 stores ----
    float4* o4 = reinterpret_cast<float4*>(out) + (size_t)row * (COLS / 4);
    #pragma unroll
    for (int k = 0; k < CHUNKS; ++k) {
        float4 v = t4[tid + k * TPB];
        float4 o;
        float vin[4] = {v.x, v.y, v.z, v.w};
        float vout[4];
        #pragma unroll
        for (int j = 0; j < 4; ++j) {
            float vx = vin[j];
            // exact searchsorted(side='left'): idx = #{ b_i < xn }, strict '>'.
            // xn = v/s uses IEEE fp32 division to match the reference bitwise.
            float xn = vx / s;
            int lo = (xn > b7) ? 8 : 0;
            float c = bnd_s[lo + 3];
            if (xn > c) lo += 4;
            c = bnd_s[lo + 1];
            if (xn > c) lo += 2;
            c = bnd_s[lo];
            if (xn > c) lo += 1;
            float q = dt_s[lo] * s;
            vout[j] = (q + vx) - vx;   // mirror q + x - stop_grad(x)
        }
        o.x = vout[0]; o.y = vout[1]; o.z = vout[2]; o.w = vout[3];
        o4[tid + k * TPB] = o;
    }
}

extern "C" void kernel_launch(void* const* d_in, const int* in_sizes, int n_in,
                              void* d_out, int out_size, void* d_ws, size_t ws_size,
                              hipStream_t stream) {
    const float* x   = (const float*)d_in[0];   // [rows, 8192] fp32
    const float* bnd = (const float*)d_in[1];   // 15 fp32 boundaries
    const float* dt  = (const float*)d_in[2];   // 16 fp32 levels
    float* out = (float*)d_out;

    int rows = in_sizes[0] / COLS;
    nf4_quant_row_kernel<<<rows, TPB, 0, stream>>>(x, bnd, dt, out);
}